// TreeLSTM_5978594476275
// MI455X (gfx1250) — compile-verified
//
#include <hip/hip_runtime.h>
#include <math.h>
#include <stdint.h>

#define SIGMA 5.0f
#define D 2048
#define H 2048
#define K0 16
#define K1 8

#define APAD 4                 // LDS row padding (floats): stride 2052 -> bank-conflict-free
#define ASTRIDE (H + APAD)

typedef __attribute__((ext_vector_type(2))) float v2f;
typedef __attribute__((ext_vector_type(8))) float v8f;

// workspace layout (float offsets)
#define WS_SQ      0                 // 16
#define WS_ASSIGN  16                // 16
#define WS_LOGITS  32                // 16*8
#define WS_GATE    160               // 16*8
#define WS_CLEAF   512               // 16*2048
#define WS_CNL     (512 + K0 * H)    // 8*2048

// ---------------------------------------------------------------------------
// 1) squared distances ||x - c_k||^2, one block per k
// ---------------------------------------------------------------------------
__global__ void k_sqdist(const float* __restrict__ x,
                         const float* __restrict__ cc, float* __restrict__ ws) {
  const int k = blockIdx.x;
  const int t = threadIdx.x;
  float acc = 0.f;
  for (int d = t; d < D; d += 256) {
    float df = x[d] - cc[k * D + d];
    acc = fmaf(df, df, acc);
  }
  __shared__ float red[256];
  red[t] = acc;
  __syncthreads();
  for (int s = 128; s > 0; s >>= 1) {
    if (t < s) red[t] += red[t + s];
    __syncthreads();
  }
  if (t == 0) ws[WS_SQ + k] = red[0];
}

// ---------------------------------------------------------------------------
// 2) assign = softmax(-sq / (2*sigma)) over 16 entries, single wave32
// ---------------------------------------------------------------------------
__global__ void k_assign(float* __restrict__ ws) {
  const int lane = threadIdx.x;
  float t = (lane < K0) ? (-ws[WS_SQ + lane] / (2.f * SIGMA)) : -1e30f;
  float m = t;
  for (int off = 16; off >= 1; off >>= 1) m = fmaxf(m, __shfl_xor(m, off, 32));
  float e = (lane < K0) ? __expf(t - m) : 0.f;
  float s = e;
  for (int off = 16; off >= 1; off >>= 1) s += __shfl_xor(s, off, 32);
  if (lane < K0) ws[WS_ASSIGN + lane] = e / s;
}

// ---------------------------------------------------------------------------
// 3) c_leaf[k,h] = assign[k] * tanh(sum_d x[d]*W[k,d,h] + b[k,h])
//    256 MB single-use stream: non-temporal loads + prefetch ahead.
//    grid (K0, H/256), block 256; one output column per thread.
// ---------------------------------------------------------------------------
__global__ void k_cleaf(const float* __restrict__ x, const float* __restrict__ w,
                        const float* __restrict__ b, float* __restrict__ ws) {
  const int k = blockIdx.x;
  const int h = blockIdx.y * 256 + threadIdx.x;
  const float* wk = w + (size_t)k * D * H + h;
  float acc = 0.f;
  for (int d0 = 0; d0 < D; d0 += 8) {
    int pd = d0 + 64 < D ? d0 + 64 : D - 1;
    __builtin_prefetch(wk + (size_t)pd * H, 0, 1);  // global_prefetch_b8
#pragma unroll
    for (int dd = 0; dd < 8; ++dd)
      acc = fmaf(x[d0 + dd],
                 __builtin_nontemporal_load(wk + (size_t)(d0 + dd) * H), acc);
  }
  float val = tanhf(acc + b[k * H + h]);
  ws[WS_CLEAF + k * H + h] = ws[WS_ASSIGN + k] * val;
}

// ---------------------------------------------------------------------------
// 4) gate_logits[k,j] = dot(c_leaf[k], W_i[j]) + b_i[j]; grid (K0,K1)
// ---------------------------------------------------------------------------
__global__ void k_gatelogits(const float* __restrict__ wi,
                             const float* __restrict__ bi, float* __restrict__ ws) {
  const int k = blockIdx.x, j = blockIdx.y, t = threadIdx.x;
  float acc = 0.f;
  for (int h = t; h < H; h += 256)
    acc = fmaf(ws[WS_CLEAF + k * H + h], wi[j * H + h], acc);
  __shared__ float red[256];
  red[t] = acc;
  __syncthreads();
  for (int s = 128; s > 0; s >>= 1) {
    if (t < s) red[t] += red[t + s];
    __syncthreads();
  }
  if (t == 0) ws[WS_LOGITS + k * K1 + j] = red[0] + bi[j];
}

// ---------------------------------------------------------------------------
// 5) gate = softmax over j (8) per k (16); single wave, lane = k
// ---------------------------------------------------------------------------
__global__ void k_gate(float* __restrict__ ws) {
  const int k = threadIdx.x;
  if (k >= K0) return;
  float v[K1];
  float m = -1e30f;
  for (int j = 0; j < K1; ++j) {
    v[j] = ws[WS_LOGITS + k * K1 + j];
    m = fmaxf(m, v[j]);
  }
  float s = 0.f;
  for (int j = 0; j < K1; ++j) {
    v[j] = __expf(v[j] - m);
    s += v[j];
  }
  for (int j = 0; j < K1; ++j) ws[WS_GATE + k * K1 + j] = v[j] / s;
}

// ---------------------------------------------------------------------------
// 6) WMMA: per j, U = C_leaf(16x2048) @ W_u[j](2048x2048), then
//    c_no_leaf[j,g] = sum_k gate[k,j] * tanh(U[k,g] + b_u[j,g]).
//    grid (K1, H/128), block 256 (8 waves x 16 output columns each).
//    A (c_leaf, 128KB) is staged into LDS once per block with the CDNA5
//    async copy path (GLOBAL_LOAD_ASYNC_TO_LDS_B128 + s_wait_asynccnt),
//    row-padded so the per-lane b64 A reads are LDS-bank-conflict-free.
//    B (single-use 128MB weight stream) uses non-temporal loads.
// ---------------------------------------------------------------------------
__global__ void k_cnoleaf_wmma(const float* __restrict__ wu,
                               const float* __restrict__ bu,
                               float* __restrict__ ws) {
  __shared__ float shA[K0 * ASTRIDE];

  const int j = blockIdx.x;
  const int tid = threadIdx.x;
  const int wave = tid >> 5;
  const int lane = tid & 31;
  const int m = lane & 15;
  const int hi = lane >> 4;
  const int n = blockIdx.y * 128 + wave * 16;

  const float* __restrict__ cl = ws + WS_CLEAF;

  // --- async stage of A into LDS: 8192 x b128 chunks, 32 per thread -------
  for (int c = tid; c < (K0 * H) / 4; c += 256) {
    const int row = c >> 9;      // / (H/4)
    const int col4 = c & 511;    // float4 index within row
    unsigned laddr = (unsigned)(uintptr_t)(&shA[row * ASTRIDE + col4 * 4]);
    uint64_t gaddr = (uint64_t)(uintptr_t)(cl + (size_t)row * H + col4 * 4);
    asm volatile("global_load_async_to_lds_b128 %0, %1, off"
                 :
                 : "v"(laddr), "v"(gaddr)
                 : "memory");
  }
  asm volatile("s_wait_asynccnt 0x0" ::: "memory");
  __syncthreads();

  const float* __restrict__ aLds = &shA[m * ASTRIDE + 2 * hi];
  const float* __restrict__ bptr =
      wu + (size_t)j * H * H + (size_t)(2 * hi) * H + n + m;

  v8f acc = {};
  for (int kk = 0; kk < H; kk += 4) {
    v2f a = *(const v2f*)(aLds + kk);            // ds_load_b64, conflict-free
    v2f b;
    b.x = __builtin_nontemporal_load(bptr + (size_t)kk * H);      // row kk+2hi
    b.y = __builtin_nontemporal_load(bptr + (size_t)kk * H + H);  // row kk+2hi+1
    acc = __builtin_amdgcn_wmma_f32_16x16x4_f32(false, a, false, b, (short)0,
                                                acc, false, false);
  }

  const float bias = bu[j * H + n + m];
  float partial = 0.f;
#pragma unroll
  for (int r = 0; r < 8; ++r) {
    int M = r + 8 * hi;
    partial = fmaf(ws[WS_GATE + M * K1 + j], tanhf(acc[r] + bias), partial);
  }
  partial += __shfl_xor(partial, 16, 32);        // combine M=0..7 with M=8..15
  if (hi == 0) ws[WS_CNL + j * H + n + m] = partial;
}

// ---------------------------------------------------------------------------
// 7) WMMA root: (8x2048 zero-padded to 16) @ root_W(2048x2048);
//    root_c[g] = sum_{k1<8} tanh(U[k1,g] + b_r[g]).  Lanes 0-15 of the C
//    layout hold exactly M=0..7, so the valid-row sum is per-lane local.
//    grid (H/128), block 256. Output duplicated (tuple of two).
// ---------------------------------------------------------------------------
__global__ void k_root_wmma(const float* __restrict__ wr,
                            const float* __restrict__ br,
                            const float* __restrict__ ws,
                            float* __restrict__ out) {
  const int wave = threadIdx.x >> 5;
  const int lane = threadIdx.x & 31;
  const int m = lane & 15;
  const int hi = lane >> 4;
  const int n = blockIdx.x * 128 + wave * 16;

  const float* __restrict__ cnl = ws + WS_CNL;
  const float* __restrict__ bptr = wr + (size_t)(2 * hi) * H + n + m;

  v8f acc = {};
  for (int kk = 0; kk < H; kk += 4) {
    v2f a = {0.f, 0.f};
    if (m < K1) a = *(const v2f*)(cnl + m * H + kk + 2 * hi);  // rows 8..15 = 0
    v2f b;
    b.x = __builtin_nontemporal_load(bptr + (size_t)kk * H);
    b.y = __builtin_nontemporal_load(bptr + (size_t)kk * H + H);
    acc = __builtin_amdgcn_wmma_f32_16x16x4_f32(false, a, false, b, (short)0,
                                                acc, false, false);
  }

  if (hi == 0) {                                  // lanes 0-15: M = 0..7 valid
    const float bias = br[n + m];
    float s = 0.f;
#pragma unroll
    for (int r = 0; r < 8; ++r) s += tanhf(acc[r] + bias);
    out[n + m] = s;
    out[H + n + m] = s;                           // second tuple element
  }
}

// ---------------------------------------------------------------------------
extern "C" void kernel_launch(void* const* d_in, const int* in_sizes, int n_in,
                              void* d_out, int out_size, void* d_ws,
                              size_t ws_size, hipStream_t stream) {
  const float* x  = (const float*)d_in[0];  // (1, D)
  const float* cc = (const float*)d_in[1];  // (K0, D)
  const float* lw = (const float*)d_in[2];  // (K0, D, H)
  const float* lb = (const float*)d_in[3];  // (K0, H)
  const float* wi = (const float*)d_in[4];  // (K1, H)
  const float* bi = (const float*)d_in[5];  // (K1,)
  const float* wu = (const float*)d_in[6];  // (K1, H, H)
  const float* bu = (const float*)d_in[7];  // (K1, H)
  const float* wr = (const float*)d_in[8];  // (H, H)
  const float* br = (const float*)d_in[9];  // (1, H)
  float* out = (float*)d_out;               // 2 * H
  float* ws  = (float*)d_ws;

  k_sqdist<<<K0, 256, 0, stream>>>(x, cc, ws);
  k_assign<<<1, 32, 0, stream>>>(ws);
  k_cleaf<<<dim3(K0, H / 256), 256, 0, stream>>>(x, lw, lb, ws);
  k_gatelogits<<<dim3(K0, K1), 256, 0, stream>>>(wi, bi, ws);
  k_gate<<<1, 32, 0, stream>>>(ws);
  k_cnoleaf_wmma<<<dim3(K1, H / 128), 256, 0, stream>>>(wu, bu, ws);
  k_root_wmma<<<H / 128, 256, 0, stream>>>(wr, br, ws, out);
}